// SmoothDCGLoss_46918222742215
// MI455X (gfx1250) — compile-verified
//
#include <hip/hip_runtime.h>
#include <hip/hip_bf16.h>

#define TOPK 20

// ---------- hardware tanh (v_tanh_f32) on every device path ----------
#if defined(__HIP_DEVICE_COMPILE__)
#  if defined(__has_builtin) && __has_builtin(__builtin_amdgcn_tanhf)
#    define FAST_TANH(x) __builtin_amdgcn_tanhf(x)
#  elif defined(__has_builtin) && __has_builtin(__builtin_amdgcn_tanh_f32)
#    define FAST_TANH(x) __builtin_amdgcn_tanh_f32(x)
#  else
// Inline-asm fallback still emits v_tanh_f32. Trailing v_nop satisfies the
// TRANS-op result hazard (ISA 7.4: 1 independent op / V_NOP before use),
// which the compiler cannot insert around opaque asm.
__device__ __forceinline__ float fast_tanh_asm(float x) {
  float r;
  asm volatile("v_tanh_f32 %0, %1\n\tv_nop" : "=v"(r) : "v"(x));
  return r;
}
#    define FAST_TANH(x) fast_tanh_asm(x)
#  endif
#else
#  define FAST_TANH(x) tanhf(x)   /* host-pass parsing only */
#endif

__device__ __forceinline__ float fast_rcp(float x) {
#if defined(__HIP_DEVICE_COMPILE__)
  return __builtin_amdgcn_rcpf(x);   // v_rcp_f32, 1 ulp — no IEEE div expansion
#else
  return 1.0f / x;
#endif
}

typedef __attribute__((ext_vector_type(2))) float v2f;
typedef __attribute__((ext_vector_type(8))) float v8f;

#if defined(__HIP_DEVICE_COMPILE__) && defined(__has_builtin)
#  if __has_builtin(__builtin_amdgcn_wmma_f32_16x16x4_f32)
#    define USE_WMMA_CUMSUM 1
#  endif
#endif

// =====================================================================
// Kernel 1: ranks[b][j] = sum_i sigmoid(scores[b,i] - top[b,j]) + 0.5
//   sigmoid(z) = 0.5 + 0.5*tanh(z/2)  ->  ranks = 0.5*N + 0.5*sum(tanh) + 0.5
// Hot loop per element-j pair: v_sub + v_tanh_f32 + v_add (1 TRANS, 2 VALU),
// 20 independent accumulator chains for ILP. Deterministic reduction.
// =====================================================================
__global__ __launch_bounds__(256) void ranks_kernel(
    const float* __restrict__ scores_top,
    const float* __restrict__ scores,
    float* __restrict__ ranks, int N) {
  const int b   = blockIdx.x;
  const int tid = threadIdx.x;

  __shared__ float tops[TOPK];
  __shared__ float part[TOPK][8];   // per-wave partials, fixed-order final sum

  if (tid < TOPK) tops[tid] = scores_top[b * TOPK + tid];
  __syncthreads();

  float th[TOPK], acc[TOPK];
#pragma unroll
  for (int j = 0; j < TOPK; ++j) { th[j] = tops[j] * 0.5f; acc[j] = 0.0f; }

  const float* row = scores + (size_t)b * N;
  const float4* s4 = reinterpret_cast<const float4*>(row);
  const int nvec = N >> 2;

  for (int i = tid; i < nvec; i += 256) {
    if (i + 512 < nvec) __builtin_prefetch(&s4[i + 512], 0, 0);  // global_prefetch_b8
    float4 s = s4[i];
    const float h0 = s.x * 0.5f, h1 = s.y * 0.5f, h2 = s.z * 0.5f, h3 = s.w * 0.5f;
#pragma unroll
    for (int j = 0; j < TOPK; ++j) {
      acc[j] += FAST_TANH(h0 - th[j]);
      acc[j] += FAST_TANH(h1 - th[j]);
      acc[j] += FAST_TANH(h2 - th[j]);
      acc[j] += FAST_TANH(h3 - th[j]);
    }
  }
  for (int i = (nvec << 2) + tid; i < N; i += 256) {   // tail (N%4)
    float h = row[i] * 0.5f;
#pragma unroll
    for (int j = 0; j < TOPK; ++j) acc[j] += FAST_TANH(h - th[j]);
  }

  // wave32 shuffle reduction
#pragma unroll
  for (int j = 0; j < TOPK; ++j) {
#pragma unroll
    for (int off = 16; off >= 1; off >>= 1)
      acc[j] += __shfl_xor(acc[j], off, 32);
  }
  const int wave = tid >> 5;
  if ((tid & 31) == 0) {
#pragma unroll
    for (int j = 0; j < TOPK; ++j) part[j][wave] = acc[j];
  }
  __syncthreads();
  if (tid < TOPK) {
    float s = 0.0f;
#pragma unroll
    for (int w = 0; w < 8; ++w) s += part[tid][w];   // fixed order: deterministic
    ranks[b * TOPK + tid] = 0.5f * (float)N + 0.5f * s + 0.5f;
  }
}

// =====================================================================
// Kernel 2: dg = labels * rcp(log2(ranks+1));  dcg = dg @ U (upper-tri ones)
// via V_WMMA_F32_16X16X4_F32 (K=20 = 5 chained k=4 steps, 2 column tiles);
// ndcg = dcg * rcp(idcg[min(ksum,p)-1]) (with JAX -1 wraparound).
// One wave per 16 batch rows; EXEC all-ones at every WMMA.
// =====================================================================
__global__ __launch_bounds__(32) void ndcg_kernel(
    const float* __restrict__ labels,
    const float* __restrict__ ranks,
    float* __restrict__ out) {
  const int lane = threadIdx.x;
  const int b0   = blockIdx.x * 16;

  __shared__ float idcg_s[TOPK];
  __shared__ int   ksum_s[16];

  if (lane < TOPK) {                      // idcg prefix vector
    float s = 0.0f;
    for (int i = 0; i <= lane; ++i) s += fast_rcp(__log2f((float)(i + 2)));
    idcg_s[lane] = s;
  }
  if (lane < 16) {                        // per-row label sums
    const float* lr = labels + (size_t)(b0 + lane) * TOPK;
    float s = 0.0f;
#pragma unroll
    for (int j = 0; j < TOPK; ++j) s += lr[j];
    ksum_s[lane] = (int)s;
  }
  __syncthreads();

  const int row   = lane & 15;    // M index this lane holds (A) / N index (B,C)
  const int khalf = lane >> 4;    // 0: K={0,1}, 1: K={2,3} within a k=4 step

  // A-matrix fragments (16x4 f32 layout), dg values; K dim = 20 exactly.
  v2f A[5];
#pragma unroll
  for (int kk = 0; kk < 5; ++kk) {
    const int    k0   = 4 * kk + 2 * khalf;
    const size_t base = (size_t)(b0 + row) * TOPK;
    A[kk].x = labels[base + k0]     * fast_rcp(__log2f(ranks[base + k0]     + 1.0f));
    A[kk].y = labels[base + k0 + 1] * fast_rcp(__log2f(ranks[base + k0 + 1] + 1.0f));
  }

#pragma unroll
  for (int ct = 0; ct < 2; ++ct) {
    const int p = 16 * ct + row;          // output column (rank position - 1)
    v8f c = {0.f, 0.f, 0.f, 0.f, 0.f, 0.f, 0.f, 0.f};

#ifdef USE_WMMA_CUMSUM
#pragma unroll
    for (int kk = 0; kk < 5; ++kk) {
      const int k0 = 4 * kk + 2 * khalf;
      v2f bm;                              // U[j,p] = (j <= p), zero-padded cols
      bm.x = (p < TOPK && k0     <= p) ? 1.0f : 0.0f;
      bm.y = (p < TOPK && k0 + 1 <= p) ? 1.0f : 0.0f;
      c = __builtin_amdgcn_wmma_f32_16x16x4_f32(
              false, A[kk], false, bm, (short)0, c, false, false);
    }
#else
    // scalar fallback (host-pass parsing / builtin-missing safety net)
#pragma unroll
    for (int v = 0; v < 8; ++v) {
      const int m = v + 8 * khalf;
      float s = 0.0f;
      if (p < TOPK) {
        const size_t base = (size_t)(b0 + m) * TOPK;
        for (int j = 0; j <= p; ++j)
          s += labels[base + j] * fast_rcp(__log2f(ranks[base + j] + 1.0f));
      }
      c[v] = s;
    }
#endif

    // D layout: VGPR v, lanes 0-15 -> M=v, lanes 16-31 -> M=v+8; N = lane&15.
#pragma unroll
    for (int v = 0; v < 8; ++v) {
      const int m = v + 8 * khalf;
      if (p < TOPK) {
        const int ks  = ksum_s[m];
        int idx = ((ks < p + 1) ? ks : (p + 1)) - 1;
        if (idx < 0) idx += TOPK;          // JAX negative-index wraparound
        out[(size_t)(b0 + m) * TOPK + p] = c[v] * fast_rcp(idcg_s[idx]);
      }
    }
  }
}

extern "C" void kernel_launch(void* const* d_in, const int* in_sizes, int n_in,
                              void* d_out, int out_size, void* d_ws, size_t ws_size,
                              hipStream_t stream) {
  const float* scores_top = (const float*)d_in[0];   // (B, 20)
  const float* scores     = (const float*)d_in[1];   // (B, N)
  const float* labels     = (const float*)d_in[2];   // (B, 20)
  float* out = (float*)d_out;                        // (B, 20)

  const int B = in_sizes[0] / TOPK;
  const int N = in_sizes[1] / B;
  float* ranks = (float*)d_ws;                       // B*20 floats of scratch

  ranks_kernel<<<B, 256, 0, stream>>>(scores_top, scores, ranks, N);
  ndcg_kernel<<<B / 16, 32, 0, stream>>>(labels, ranks, out);
}